// SparseLinear_47880295416581
// MI455X (gfx1250) — compile-verified
//
#include <hip/hip_runtime.h>

typedef float v2f __attribute__((ext_vector_type(2)));
typedef float v8f __attribute__((ext_vector_type(8)));
typedef int   vb128 __attribute__((vector_size(16)));   // matches builtin param type

#define GLOBAL_AS __attribute__((address_space(1)))
#define LDS_AS    __attribute__((address_space(3)))

#define B_DIM   8
#define S_DIM   2048
#define M_DIM   2048                 // contraction dim (K of the GEMM)
#define N_DIM   2048
#define K_NNZ   205
#define T_DIM   (B_DIM * S_DIM)      // 16384 token positions
#define KC      256                  // K-chunk staged in LDS
#define LDSP    (KC + 4)             // 260: +4 floats -> A-frag lanes on distinct bank pairs
#define BUFELTS (128 * LDSP)         // one LDS buffer (floats)
#define NCHUNK  (M_DIM / KC)         // 8

// ---------------------------------------------------------------------------
// Kernel 1: zero the packed dense W^T buffer (M*N f32 = 16 MiB in d_ws)
// ---------------------------------------------------------------------------
__global__ void zero_wt_kernel(float4* __restrict__ wt) {
  size_t i = (size_t)blockIdx.x * blockDim.x + threadIdx.x;
  wt[i] = make_float4(0.f, 0.f, 0.f, 0.f);
}

// ---------------------------------------------------------------------------
// Kernel 2: scatter CSR values into packed dense W^T.
// Element (k=m, n) lives at wt[(m>>1)*2N + 2n + (m&1)], so a WMMA B-fragment
// lane reads rows {k, k+1} at one column as ONE global_load_b64.
// ---------------------------------------------------------------------------
__global__ void scatter_wt_kernel(const float* __restrict__ wval,
                                  const int* __restrict__ cols,
                                  float* __restrict__ wt) {
  int i = blockIdx.x * blockDim.x + threadIdx.x;
  if (i >= N_DIM * K_NNZ) return;
  int n = i / K_NNZ;                 // uniform CSR: row id by division
  int m = cols[i];
  wt[(size_t)(m >> 1) * (2 * N_DIM) + 2 * n + (m & 1)] = wval[i];
}

// ---------------------------------------------------------------------------
// Kernel 3: Y[T,N] = X[T,M] @ W^T + bias via V_WMMA_F32_16X16X4_F32.
// 1024 threads = 32 wave32. Block tile 128t x 256n; wave = 16t x 64n (4 accs).
// x chunks staged via GLOBAL_LOAD_ASYNC_TO_LDS_B128 into double-buffered LDS
// (no VGPR staging -> no spills); copy of chunk k+1 overlaps WMMA on chunk k.
// ---------------------------------------------------------------------------
__global__ __launch_bounds__(1024) void spmm_wmma_kernel(
    const float* __restrict__ x, const float* __restrict__ wt,
    const float* __restrict__ bias, float* __restrict__ out) {
  __shared__ float lds_x[2 * BUFELTS];         // 266,240 B (<= 320 KB/WGP)

  const int tid  = threadIdx.x;
  const int t0   = blockIdx.x * 128;
  const int nblk = blockIdx.y * 256;

  const int lane = tid & 31;
  const int wid  = tid >> 5;                   // wave 0..31
  const int lrow = lane & 15;                  // fragment row/col lane index
  const int lhi  = lane >> 4;                  // K-pair select (ISA 7.12.2)
  const int tsub = wid & 7;                    // t-subtile 0..7 (16 rows each)
  const int nsub = wid >> 3;                   // n-subtile 0..3 (64 cols each)
  const int n0   = nblk + nsub * 64;

  v8f acc0 = {}, acc1 = {}, acc2 = {}, acc3 = {};

  // this thread's 8 (row, col4) slots of each 128x256 chunk
  const int r0  = tid >> 6;                    // rows r0, r0+16, ..., r0+112
  const int c4  = tid & 63;                    // 64 float4 per chunk-row

  // ---- async copy issue: chunk kc -> LDS buffer b ----
  #define ISSUE_CHUNK(kc_, b_)                                                   \
    {                                                                            \
      _Pragma("unroll")                                                          \
      for (int j = 0; j < 8; ++j) {                                              \
        int r = r0 + j * 16;                                                     \
        const float* g = x + (size_t)(t0 + r) * M_DIM + (kc_) * KC + c4 * 4;     \
        float* l = &lds_x[(b_) * BUFELTS + r * LDSP + c4 * 4];                   \
        __builtin_amdgcn_global_load_async_to_lds_b128(                          \
            (GLOBAL_AS vb128*)g, (LDS_AS vb128*)l, 0, 0);                        \
      }                                                                          \
    }

  ISSUE_CHUNK(0, 0);

  // A fragment source (LDS): lane holds A[trow, kk+2*lhi .. +1] -> b64
  const float* aw0 = &lds_x[(tsub * 16 + lrow) * LDSP + 2 * lhi];
  // B fragment base (global, packed layout)
  const float* bbase = wt + (size_t)lhi * (2 * N_DIM) + 2 * (n0 + lrow);

  for (int kc = 0; kc < NCHUNK; ++kc) {
    asm volatile("s_wait_asynccnt 0x0" ::: "memory");  // own async writes done
    __syncthreads();                                   // whole chunk visible

    if (kc + 1 < NCHUNK) ISSUE_CHUNK(kc + 1, (kc + 1) & 1);

    const float* aw = aw0 + (kc & 1) * BUFELTS;
    const float* bp = bbase + (size_t)kc * KC * N_DIM; // == (kc*KC/2)*2N
    #pragma unroll 8
    for (int kk = 0; kk < KC; kk += 4) {
      v2f a = *(const v2f*)(aw + kk);
      size_t boff = (size_t)kk * N_DIM;                // == (kk/2)*2N
      v2f vb0 = *(const v2f*)(bp + boff);
      v2f vb1 = *(const v2f*)(bp + boff + 32);         // +16 cols
      v2f vb2 = *(const v2f*)(bp + boff + 64);
      v2f vb3 = *(const v2f*)(bp + boff + 96);
      acc0 = __builtin_amdgcn_wmma_f32_16x16x4_f32(false, a, false, vb0, (short)0, acc0, false, false);
      acc1 = __builtin_amdgcn_wmma_f32_16x16x4_f32(false, a, false, vb1, (short)0, acc1, false, false);
      acc2 = __builtin_amdgcn_wmma_f32_16x16x4_f32(false, a, false, vb2, (short)0, acc2, false, false);
      acc3 = __builtin_amdgcn_wmma_f32_16x16x4_f32(false, a, false, vb3, (short)0, acc3, false, false);
    }
  }

  // ---- store 16x16 C/D tiles (+ bias): VGPR r -> out row base + r + 8*lhi ----
  float* obase = out + (size_t)(t0 + tsub * 16 + 8 * lhi) * N_DIM;
  int col; float bv;
  col = n0 + 0 * 16 + lrow;  bv = bias[col];
  #pragma unroll
  for (int r = 0; r < 8; ++r) obase[(size_t)r * N_DIM + col] = acc0[r] + bv;
  col = n0 + 1 * 16 + lrow;  bv = bias[col];
  #pragma unroll
  for (int r = 0; r < 8; ++r) obase[(size_t)r * N_DIM + col] = acc1[r] + bv;
  col = n0 + 2 * 16 + lrow;  bv = bias[col];
  #pragma unroll
  for (int r = 0; r < 8; ++r) obase[(size_t)r * N_DIM + col] = acc2[r] + bv;
  col = n0 + 3 * 16 + lrow;  bv = bias[col];
  #pragma unroll
  for (int r = 0; r < 8; ++r) obase[(size_t)r * N_DIM + col] = acc3[r] + bv;
}

// ---------------------------------------------------------------------------
extern "C" void kernel_launch(void* const* d_in, const int* in_sizes, int n_in,
                              void* d_out, int out_size, void* d_ws, size_t ws_size,
                              hipStream_t stream) {
  const float* x    = (const float*)d_in[0];
  const float* wval = (const float*)d_in[1];
  const float* bias = (const float*)d_in[2];
  const int*   cols = (const int*)d_in[3];
  // d_in[4] = indptr: uniform (n*K_NNZ), recomputed by division in scatter.

  float* wt  = (float*)d_ws;     // packed dense W^T: M*N f32 = 16 MiB (L2-resident)
  float* out = (float*)d_out;

  // 1) zero W^T  (M*N/4 float4 = 1,048,576 -> exact multiple of 256)
  zero_wt_kernel<<<(M_DIM * N_DIM / 4) / 256, 256, 0, stream>>>((float4*)wt);
  // 2) scatter nnz values
  scatter_wt_kernel<<<(N_DIM * K_NNZ + 255) / 256, 256, 0, stream>>>(wval, cols, wt);
  // 3) WMMA GEMM + bias: grid = (T/128, N/256)
  dim3 grid(T_DIM / 128, N_DIM / 256);
  spmm_wmma_kernel<<<grid, 1024, 0, stream>>>(x, wt, bias, out);
}